// VectorizedQuantumCircuit_90142773609090
// MI455X (gfx1250) — compile-verified
//
#include <hip/hip_runtime.h>

// ---------------------------------------------------------------------------
// VectorizedQuantumCircuit on MI455X (gfx1250, wave32).
//
// The circuit is real-valued: state = 4096 floats per batch element, kept
// entirely in registers (128 VGPRs/lane * 32 lanes). Layout: d = r*32 + lane,
// so d-bits 0..4 live in the lane index and d-bits 5..11 live in the register
// index r.  All loops are fully unrolled so v[128] never spills.
//
// Readout einsum probs x SIGNS is done with V_WMMA_F32_16X16X4_F32 across the
// 16 waves (= 16 batch rows) of a block, staged through padded LDS tiles.
// ---------------------------------------------------------------------------

#define NQ     12
#define RPL    128          // state registers per lane (4096 / 32)
#define WAVES  16           // waves (= batch rows) per block
#define BLOCK  (WAVES * 32)
#define TILE_STRIDE 132     // 128 + 4 pad -> conflict-free A-fragment loads

typedef __attribute__((ext_vector_type(2))) float v2f;
typedef __attribute__((ext_vector_type(8))) float v8f;

#if __has_builtin(__builtin_amdgcn_wmma_f32_16x16x4_f32)
#define HAVE_WMMA_F32X4 1
#endif

__global__ __launch_bounds__(BLOCK) void qc_kernel(const float* __restrict__ x,
                                                   const float* __restrict__ theta,
                                                   float* __restrict__ out)
{
#ifdef HAVE_WMMA_F32X4
    __shared__ float tileP[WAVES][TILE_STRIDE];   // 16 rows x 128 K (+pad)
    __shared__ float cred[WAVES * 32 * 8];        // 16 partial C matrices
#endif

    const int tid  = threadIdx.x;
    const int lane = tid & 31;
    const int wave = tid >> 5;
    const int b    = blockIdx.x * WAVES + wave;

    // ---- encoding angles -------------------------------------------------
    float cx[NQ], sx[NQ];
    #pragma unroll
    for (int q = 0; q < NQ; ++q) {
        float a = 0.5f * x[b * NQ + q];
        cx[q] = cosf(a);
        sx[q] = sinf(a);
    }

    // lane-bit prefix (qubits 0..4)
    float pl = 1.0f;
    #pragma unroll
    for (int q = 0; q < 5; ++q)
        pl *= ((lane >> q) & 1) ? sx[q] : cx[q];

    // product-state build over in-lane qubits 5..11 (doubling, all indices
    // compile-time after unroll -> pure registers)
    float v[RPL];
    v[0] = pl;
    #pragma unroll
    for (int rb = 0; rb < 7; ++rb) {
        #pragma unroll
        for (int t = 0; t < (1 << rb); ++t) {
            v[t | (1 << rb)] = v[t] * sx[5 + rb];
            v[t]             = v[t] * cx[5 + rb];
        }
    }

    // ---- 4 variational layers -------------------------------------------
    #pragma unroll
    for (int layer = 0; layer < 4; ++layer) {
        float ct[NQ], st[NQ];
        #pragma unroll
        for (int q = 0; q < NQ; ++q) {
            float a = 0.5f * theta[layer * NQ + q];
            ct[q] = cosf(a);
            st[q] = sinf(a);
        }

        // RY on cross-lane qubits 0..4: shuffle + FMA
        #pragma unroll
        for (int q = 0; q < 5; ++q) {
            const float c  = ct[q];
            const float ss = ((lane >> q) & 1) ? st[q] : -st[q];
            #pragma unroll
            for (int r = 0; r < RPL; ++r) {
                float p = __shfl_xor(v[r], 1 << q, 32);
                v[r] = fmaf(ss, p, c * v[r]);
            }
        }

        // RY on in-lane qubits 5..11: register butterflies
        #pragma unroll
        for (int rb = 0; rb < 7; ++rb) {
            const float c = ct[5 + rb], s = st[5 + rb];
            #pragma unroll
            for (int h = 0; h < 64; ++h) {
                int i = ((h >> rb) << (rb + 1)) | (h & ((1 << rb) - 1));
                int j = i | (1 << rb);
                float a0 = v[i], a1 = v[j];
                v[i] = c * a0 - s * a1;
                v[j] = fmaf(s, a0, c * a1);
            }
        }

        // Ring of CNOTs, applied in order (0,1),(1,2),...,(11,0).
        // (0,1)..(3,4): pure lane-bit permutations -> fuse into ONE gather.
        // src(l) = pi01(pi12(pi23(pi34(l))))
        {
            int sl = lane;
            sl = ((sl >> 3) & 1) ? (sl ^ 16) : sl;  // pi34: bit3 controls bit4
            sl = ((sl >> 2) & 1) ? (sl ^ 8)  : sl;  // pi23
            sl = ((sl >> 1) & 1) ? (sl ^ 4)  : sl;  // pi12
            sl = ( sl        & 1) ? (sl ^ 2) : sl;  // pi01
            #pragma unroll
            for (int r = 0; r < RPL; ++r)
                v[r] = __shfl(v[r], sl, 32);
        }

        // CNOT(4,5): lane-bit4 controls swap of r-bit0 pairs (v_cndmask)
        {
            const bool cond = (lane >> 4) & 1;
            #pragma unroll
            for (int h = 0; h < 64; ++h) {
                int i = 2 * h, j = i + 1;
                float a0 = v[i], a1 = v[j];
                v[i] = cond ? a1 : a0;
                v[j] = cond ? a0 : a1;
            }
        }

        // CNOT(5,6)..(10,11): both bits in-register -> free renames
        #pragma unroll
        for (int cb = 0; cb < 6; ++cb) {
            #pragma unroll
            for (int r = 0; r < RPL; ++r) {
                if (((r >> cb) & 1) && !((r >> (cb + 1)) & 1)) {
                    int j = r | (1 << (cb + 1));
                    float tmpv = v[r]; v[r] = v[j]; v[j] = tmpv;
                }
            }
        }

        // CNOT(11,0): r-bit6 controls cross-lane swap over lane-bit0
        #pragma unroll
        for (int r = 64; r < RPL; ++r)
            v[r] = __shfl_xor(v[r], 1, 32);
    }

    // ---- readout: out[b,q] = sum_d v[d]^2 * (1 - 2*bit_q(d)) -------------
#ifdef HAVE_WMMA_F32X4
    // GEMM via V_WMMA_F32_16X16X4_F32: A = probs tile (16 rows x 4 K),
    // B = +/-1 sign matrix (4 K x 16 qubits, cols 12..15 zero-padded).
    v8f acc = {};
    const int row   = lane & 15;   // A: M = lane%16
    const int khalf = lane >> 4;   // lane group picks K pair {0,1} or {2,3}
    const int nq    = lane & 15;   // B: N = lane%16

    #pragma unroll
    for (int c = 0; c < 32; ++c) {
        // stage this wave's probs for K range [c*128, c*128+128)
        #pragma unroll
        for (int j = 0; j < 4; ++j) {
            float pv = v[c * 4 + j];                 // d = (c*4+j)*32 + lane
            tileP[wave][j * 32 + lane] = pv * pv;
        }
        __syncthreads();
        // each wave owns 2 K-chunks of 4 inside the 128-wide tile
        #pragma unroll
        for (int t = 0; t < 2; ++t) {
            const int kb = wave * 8 + t * 4;
            v2f a, bm;
            a.x = tileP[row][kb + 2 * khalf + 0];
            a.y = tileP[row][kb + 2 * khalf + 1];
            const int d0 = c * 128 + kb + 2 * khalf;  // global K (= d index)
            bm.x = (nq < NQ) ? (1.0f - 2.0f * (float)((d0 >> nq) & 1))       : 0.0f;
            bm.y = (nq < NQ) ? (1.0f - 2.0f * (float)(((d0 + 1) >> nq) & 1)) : 0.0f;
            acc = __builtin_amdgcn_wmma_f32_16x16x4_f32(
                      false, a, false, bm, (short)0, acc, false, false);
        }
        __syncthreads();
    }

    // cross-wave reduction of the 16 partial C matrices
    #pragma unroll
    for (int i = 0; i < 8; ++i)
        cred[(wave * 32 + lane) * 8 + i] = acc[i];
    __syncthreads();

    if (tid < 256) {
        const int l = tid >> 3;      // C-layout lane 0..31
        const int i = tid & 7;       // C-layout VGPR 0..7
        float s = 0.0f;
        #pragma unroll
        for (int w = 0; w < WAVES; ++w)
            s += cred[(w * 32 + l) * 8 + i];
        const int M = i + 8 * (l >> 4);   // batch row within block
        const int N = l & 15;             // qubit
        if (N < NQ)
            out[(blockIdx.x * WAVES + M) * NQ + N] = s;
    }
#else
    // Fallback: per-wave signed reduction (no WMMA available)
    float tot = 0.0f, dq[7] = {0, 0, 0, 0, 0, 0, 0};
    #pragma unroll
    for (int r = 0; r < RPL; ++r) {
        float pv = v[r] * v[r];
        tot += pv;
        #pragma unroll
        for (int rb = 0; rb < 7; ++rb)
            dq[rb] += ((r >> rb) & 1) ? -pv : pv;
    }
    float res[NQ];
    #pragma unroll
    for (int q = 0; q < 5; ++q)
        res[q] = ((lane >> q) & 1) ? -tot : tot;
    #pragma unroll
    for (int rb = 0; rb < 7; ++rb)
        res[5 + rb] = dq[rb];
    #pragma unroll
    for (int m = 1; m < 32; m <<= 1) {
        #pragma unroll
        for (int q = 0; q < NQ; ++q)
            res[q] += __shfl_xor(res[q], m, 32);
    }
    #pragma unroll
    for (int q = 0; q < NQ; ++q)
        if (lane == q) out[b * NQ + q] = res[q];
#endif
}

extern "C" void kernel_launch(void* const* d_in, const int* in_sizes, int n_in,
                              void* d_out, int out_size, void* d_ws, size_t ws_size,
                              hipStream_t stream) {
    const float* x     = (const float*)d_in[0];   // (8192, 12)
    const float* theta = (const float*)d_in[1];   // (4, 12)
    float* out = (float*)d_out;                   // (8192, 12)
    (void)in_sizes; (void)n_in; (void)out_size; (void)d_ws; (void)ws_size;

    dim3 grid(8192 / WAVES);   // 512 blocks
    dim3 block(BLOCK);         // 512 threads = 16 waves = 16 batch rows
    qc_kernel<<<grid, block, 0, stream>>>(x, theta, out);
}